// convGNN_85392539779308
// MI455X (gfx1250) — compile-verified
//
#include <hip/hip_runtime.h>
#include <hip/hip_bf16.h>

typedef __attribute__((ext_vector_type(2))) float v2f;
typedef __attribute__((ext_vector_type(8))) float v8f;

#define NNODES   100000
#define NFEAT    512
#define HIDDEN   128
#define NOUT     2
#define KCHUNK   64

// ---------------- degree / normalization ----------------
__global__ void deg_init_kernel(float* __restrict__ deg, int n) {
    int i = blockIdx.x * blockDim.x + threadIdx.x;
    if (i < n) deg[i] = 1.0f;  // self-loop contributes 1
}

__global__ void deg_accum_kernel(const long long* __restrict__ edge, float* __restrict__ deg,
                                 int nedges) {
    int e = blockIdx.x * blockDim.x + threadIdx.x;
    if (e < nedges) {
        int dst = (int)edge[nedges + e];
        atomicAdd(&deg[dst], 1.0f);
    }
}

__global__ void dinv_kernel(const float* __restrict__ deg, float* __restrict__ dinv, int n) {
    int i = blockIdx.x * blockDim.x + threadIdx.x;
    if (i < n) dinv[i] = rsqrtf(deg[i]);
}

// ---------------- h = x @ w1 via fp32 WMMA ----------------
// Grid: ceil(NNODES/128) blocks of 256 threads (8 waves).
// Each wave computes a 16-row x 128-col output tile; K staged via LDS in 64-chunks.
// B is stored K-pair interleaved in LDS so each lane's B fragment is one b64 load.
__global__ __launch_bounds__(256) void gemm1_wmma_kernel(const float* __restrict__ x,
                                                         const float* __restrict__ w1,
                                                         float* __restrict__ h) {
    __shared__ __align__(16) float ldsA[128 * KCHUNK];        // [row][k]            32 KB
    __shared__ __align__(16) float ldsB[(KCHUNK / 2) * 256];  // [k/2][n][k%2]       32 KB

    const int tid  = threadIdx.x;
    const int wave = tid >> 5;
    const int lane = tid & 31;
    const int lhalf = lane >> 4;   // 0: K pair (0,1) / M 0..7 ; 1: K pair (2,3) / M 8..15
    const int l16   = lane & 15;
    const int row0 = blockIdx.x * 128;

    v8f acc[8] = {};

    for (int kc = 0; kc < NFEAT; kc += KCHUNK) {
        // cooperative load A tile: 128 rows x 64 cols = 2048 float4
        #pragma unroll
        for (int i = 0; i < 8; ++i) {
            int idx = tid + i * 256;          // 0..2047
            int r   = idx >> 4;               // 16 float4 per row
            int c4  = (idx & 15) << 2;
            int grow = row0 + r;
            if (grow > NNODES - 1) grow = NNODES - 1;  // clamp (tail rows never stored)
            const float4 v = *(const float4*)(x + (size_t)grow * NFEAT + kc + c4);
            *(float4*)(&ldsA[r * KCHUNK + c4]) = v;
        }
        // cooperative load B tile: 64 rows x 128 cols, K-pair interleaved.
        // Task: row-pair p (rows 2p,2p+1), 4 columns c4..c4+3 -> 8 contiguous LDS floats.
        #pragma unroll
        for (int i = 0; i < 4; ++i) {
            int idx = tid + i * 256;          // 0..1023
            int p   = idx >> 5;               // row pair 0..31
            int c4  = (idx & 31) << 2;        // column base
            const float4 v0 = *(const float4*)(w1 + (size_t)(kc + 2 * p)     * HIDDEN + c4);
            const float4 v1 = *(const float4*)(w1 + (size_t)(kc + 2 * p + 1) * HIDDEN + c4);
            float* dp = &ldsB[p * 256 + c4 * 2];
            *(float4*)(dp)     = make_float4(v0.x, v1.x, v0.y, v1.y);
            *(float4*)(dp + 4) = make_float4(v0.z, v1.z, v0.w, v1.w);
        }
        __syncthreads();

        #pragma unroll 4
        for (int ks = 0; ks < KCHUNK / 4; ++ks) {
            // A fragment: lanes 0-15 hold K=(4ks,4ks+1), lanes 16-31 hold K=(4ks+2,4ks+3)
            const v2f a = *(const v2f*)(&ldsA[(wave * 16 + l16) * KCHUNK + ks * 4 + lhalf * 2]);
            #pragma unroll
            for (int nt = 0; nt < 8; ++nt) {
                // B fragment: one aligned b64 load of (B[k][n], B[k+1][n])
                const v2f b = *(const v2f*)(&ldsB[(ks * 2 + lhalf) * 256 + (nt * 16 + l16) * 2]);
                acc[nt] = __builtin_amdgcn_wmma_f32_16x16x4_f32(
                    false, a, false, b, (short)0, acc[nt], false, false);
            }
        }
        __syncthreads();
    }

    // store: VGPR r holds (M=r, N=lane) for lanes 0-15, (M=8+r, N=lane-16) for 16-31
    const int mbase = row0 + wave * 16 + lhalf * 8;
    #pragma unroll
    for (int nt = 0; nt < 8; ++nt) {
        #pragma unroll
        for (int r = 0; r < 8; ++r) {
            int grow = mbase + r;
            if (grow < NNODES)
                h[(size_t)grow * HIDDEN + nt * 16 + l16] = acc[nt][r];
        }
    }
}

// ---------------- agg init with self-loop term ----------------
__global__ void selfloop_kernel(const float* __restrict__ h, const float* __restrict__ dinv,
                                float* __restrict__ agg, int n) {
    int i = blockIdx.x * blockDim.x + threadIdx.x;   // over n*HIDDEN
    if (i < n * HIDDEN) {
        int node = i >> 7;
        float d = dinv[node];
        agg[i] = h[i] * d * d;
    }
}

// ---------------- edge scatter: one wave per edge ----------------
__global__ __launch_bounds__(256) void scatter_kernel(const long long* __restrict__ edge,
                                                      const float* __restrict__ h,
                                                      const float* __restrict__ dinv,
                                                      float* __restrict__ agg, int nedges) {
    int t = blockIdx.x * blockDim.x + threadIdx.x;
    int e = t >> 5;
    int lane = t & 31;
    if (e >= nedges) return;
    int src = (int)edge[e];
    int dst = (int)edge[nedges + e];
    float norm = dinv[src] * dinv[dst];
    const float4 v = *(const float4*)(h + (size_t)src * HIDDEN + lane * 4);
    float* out = agg + (size_t)dst * HIDDEN + lane * 4;
    atomicAdd(out + 0, v.x * norm);
    atomicAdd(out + 1, v.y * norm);
    atomicAdd(out + 2, v.z * norm);
    atomicAdd(out + 3, v.w * norm);
}

// ---------------- final: out = (agg + b1) @ w2 + b2, one wave per row ----------------
__global__ __launch_bounds__(256) void final_kernel(const float* __restrict__ agg,
                                                    const float* __restrict__ b1,
                                                    const float* __restrict__ w2,
                                                    const float* __restrict__ b2,
                                                    float* __restrict__ out, int n) {
    int t = blockIdx.x * blockDim.x + threadIdx.x;
    int row = t >> 5;
    int lane = t & 31;
    if (row >= n) return;
    const float4 a = *(const float4*)(agg + (size_t)row * HIDDEN + lane * 4);
    float p0 = 0.f, p1 = 0.f;
    #pragma unroll
    for (int k = 0; k < 4; ++k) {
        int f = lane * 4 + k;
        float av = ((const float*)&a)[k] + b1[f];
        p0 += av * w2[f * 2 + 0];
        p1 += av * w2[f * 2 + 1];
    }
    #pragma unroll
    for (int off = 16; off > 0; off >>= 1) {
        p0 += __shfl_down(p0, off, 32);
        p1 += __shfl_down(p1, off, 32);
    }
    if (lane == 0) {
        out[(size_t)row * 2 + 0] = p0 + b2[0];
        out[(size_t)row * 2 + 1] = p1 + b2[1];
    }
}

extern "C" void kernel_launch(void* const* d_in, const int* in_sizes, int n_in,
                              void* d_out, int out_size, void* d_ws, size_t ws_size,
                              hipStream_t stream) {
    const float* x  = (const float*)d_in[0];
    const float* w1 = (const float*)d_in[1];
    const float* b1 = (const float*)d_in[2];
    const float* w2 = (const float*)d_in[3];
    const float* b2 = (const float*)d_in[4];
    const long long* edge = (const long long*)d_in[5];

    const int n = in_sizes[0] / NFEAT;       // 100000
    const int nedges = in_sizes[5] / 2;      // 1600000

    // workspace layout (floats): deg[n] | dinv[n] | h[n*128] | agg[n*128]
    float* ws   = (float*)d_ws;
    float* deg  = ws;
    float* dinv = ws + n;
    float* h    = ws + 2 * (size_t)n;
    float* agg  = h + (size_t)n * HIDDEN;
    float* out  = (float*)d_out;

    deg_init_kernel<<<(n + 255) / 256, 256, 0, stream>>>(deg, n);
    deg_accum_kernel<<<(nedges + 255) / 256, 256, 0, stream>>>(edge, deg, nedges);
    dinv_kernel<<<(n + 255) / 256, 256, 0, stream>>>(deg, dinv, n);

    gemm1_wmma_kernel<<<(n + 127) / 128, 256, 0, stream>>>(x, w1, h);

    selfloop_kernel<<<((size_t)n * HIDDEN + 255) / 256, 256, 0, stream>>>(h, dinv, agg, n);
    scatter_kernel<<<(nedges * 32 + 255) / 256, 256, 0, stream>>>(edge, h, dinv, agg, nedges);

    final_kernel<<<(n * 32 + 255) / 256, 256, 0, stream>>>(agg, b1, w2, b2, out, n);
}